// ReadMemory_27762668601943
// MI455X (gfx1250) — compile-verified
//
#include <hip/hip_runtime.h>
#include <hip/hip_bf16.h>

// ---------------- problem constants ----------------
#define BB      64
#define MM      128
#define DMODEL  1024
#define DADDR   64
#define DMEM    128
#define HH      8
#define NPROJ   (HH * DADDR)        // 512
#define KROWS   (MM * DADDR)        // 8192 (contraction length of stage-1)
#define KSPLIT  8
#define EPSF    1e-05f

typedef __attribute__((ext_vector_type(2))) float v2f;
typedef __attribute__((ext_vector_type(8))) float v8f;

__device__ __forceinline__ float softplus_f(float x) {
    // numerically stable softplus: max(x,0) + log1p(exp(-|x|))
    return fmaxf(x, 0.0f) + log1pf(__expf(-fabsf(x)));
}

// ---------------------------------------------------------------------------
// K1: k = softplus(addresses) (M x dA), norm_k = sum_m k  (dA)
// ---------------------------------------------------------------------------
__global__ __launch_bounds__(64) void rm_kaddr_kernel(const float* __restrict__ addresses,
                                                      float* __restrict__ kbuf,
                                                      float* __restrict__ norm_k) {
    int a = threadIdx.x;            // 64 threads, one per dA column
    float s = 0.0f;
    for (int m = 0; m < MM; ++m) {
        float v = softplus_f(addresses[m * DADDR + a]);
        kbuf[m * DADDR + a] = v;
        s += v;
    }
    norm_k[a] = s;
}

// ---------------------------------------------------------------------------
// K2: aq = softplus(query @ Wa + ba), rq = softplus(query @ Wr + br)
//     64x1024 @ 1024x512 via V_WMMA_F32_16X16X4_F32.
//     grid.x = 2 * (4 Mtiles * 32 Ntiles) = 256 blocks of 1 wave (32 threads).
// A frag (16x4 f32): lane L holds row M=L&15; comps = K {0,1} (lanes 0-15)
//                    or K {2,3} (lanes 16-31).
// B frag (4x16 f32): lane L holds col N=L&15; comps = K rows {0,1} / {2,3}.
// C/D (16x16 f32, 8 VGPRs): vgpr r, lanes 0-15 -> (M=r, N=lane),
//                           lanes 16-31 -> (M=r+8, N=lane-16).
// ---------------------------------------------------------------------------
__global__ __launch_bounds__(32) void rm_proj_kernel(const float* __restrict__ query,
                                                     const float* __restrict__ Wa,
                                                     const float* __restrict__ ba,
                                                     const float* __restrict__ Wr,
                                                     const float* __restrict__ br,
                                                     float* __restrict__ aq,
                                                     float* __restrict__ rq) {
    const int tile = blockIdx.x;
    const float* W;  const float* bias;  float* out;
    if (tile < 128) { W = Wa; bias = ba; out = aq; }
    else            { W = Wr; bias = br; out = rq; }
    const int t     = tile & 127;
    const int Mbase = (t >> 5) * 16;    // 4 tiles over B=64 rows
    const int Nbase = (t & 31) * 16;    // 32 tiles over 512 cols
    const int lane  = threadIdx.x;
    const int half  = lane >> 4;        // 0: K {0,1}, 1: K {2,3}
    const int l     = lane & 15;
    const int row   = Mbase + l;
    const int col   = Nbase + l;

    v8f c = {};
    for (int kk = 0; kk < DMODEL; kk += 4) {
        const int ka = kk + half * 2;
        v2f A, Bf;
        A.x  = query[row * DMODEL + ka];
        A.y  = query[row * DMODEL + ka + 1];
        Bf.x = W[(size_t)ka * NPROJ + col];
        Bf.y = W[(size_t)(ka + 1) * NPROJ + col];
        c = __builtin_amdgcn_wmma_f32_16x16x4_f32(false, A, false, Bf,
                                                  (short)0, c, false, false);
    }
#pragma unroll
    for (int r = 0; r < 8; ++r) {
        const int mrow = Mbase + r + half * 8;
        out[mrow * NPROJ + col] = softplus_f(c[r] + bias[col]);
    }
}

// ---------------------------------------------------------------------------
// K3: per (b,h):  w[m] = aq . k[m,:],  q[m] = rq . normalizer[b,m,:],
//                 den1 = aq . norm_k + EPS
// grid = B*H blocks of 128 threads (one per m)
// ---------------------------------------------------------------------------
__global__ __launch_bounds__(128) void rm_weights_kernel(const float* __restrict__ aq,
                                                         const float* __restrict__ rq,
                                                         const float* __restrict__ kbuf,
                                                         const float* __restrict__ norm_k,
                                                         const float* __restrict__ normalizer,
                                                         float* __restrict__ w,
                                                         float* __restrict__ q,
                                                         float* __restrict__ den1) {
    const int bh = blockIdx.x;
    const int b  = bh >> 3;
    const int m  = threadIdx.x;
    const float* aqv  = aq + (size_t)bh * DADDR;   // aq[b, h*64 + a] contiguous per (b,h)
    const float* rqv  = rq + (size_t)bh * DADDR;
    const float* krow = kbuf + m * DADDR;
    const float* nrow = normalizer + ((size_t)b * MM + m) * DADDR;
    float sw = 0.0f, sq = 0.0f;
    for (int a = 0; a < DADDR; ++a) {
        sw = fmaf(aqv[a], krow[a], sw);
        sq = fmaf(rqv[a], nrow[a], sq);
    }
    w[(size_t)bh * MM + m] = sw;
    q[(size_t)bh * MM + m] = sq;
    if (m == 0) {
        float s = 0.0f;
        for (int a = 0; a < DADDR; ++a) s = fmaf(aqv[a], norm_k[a], s);
        den1[bh] = s + EPSF;
    }
}

// ---------------------------------------------------------------------------
// K4: zero the stage-1 partial accumulator (B*H*DMEM floats)
// ---------------------------------------------------------------------------
__global__ __launch_bounds__(256) void rm_zero_kernel(float* __restrict__ p, int n) {
    int i = blockIdx.x * 256 + threadIdx.x;
    if (i < n) p[i] = 0.0f;
}

// ---------------------------------------------------------------------------
// K5: MAIN (bandwidth-bound, streams all 256 MB of `matrix` exactly once)
//   num_hat[b,h,d] = sum_{k=(m,a)} (w[b,h,m]*rq[b,h,a]) * matrix[b,k,d]
// grid = (B, KSPLIT), 256 threads (8 waves). Each wave owns 16 d-columns and
// runs 256 K-steps of V_WMMA_F32_16X16X4_F32; A-fragment built on the fly
// from LDS-resident w/rq (rows 8..15 padded zero). Partial 8x128 tiles are
// accumulated with global_atomic_add_f32.
// ---------------------------------------------------------------------------
__global__ __launch_bounds__(256) void rm_stage1_kernel(const float* __restrict__ matrix,
                                                        const float* __restrict__ w,
                                                        const float* __restrict__ rq,
                                                        float* __restrict__ num_partial) {
    __shared__ float s_w[HH * MM];       // 8 x 128
    __shared__ float s_rq[HH * DADDR];   // 8 x 64

    const int b   = blockIdx.x;
    const int ks  = blockIdx.y;
    const int tid = threadIdx.x;

    for (int i = tid; i < HH * MM; i += 256)    s_w[i]  = w[(size_t)b * HH * MM + i];
    for (int i = tid; i < HH * DADDR; i += 256) s_rq[i] = rq[(size_t)b * HH * DADDR + i];
    __syncthreads();

    const int wv    = tid >> 5;          // wave id -> d-tile
    const int lane  = tid & 31;
    const int half  = lane >> 4;         // K sub-pair {0,1} vs {2,3}
    const int l     = lane & 15;
    const int dcol  = wv * 16 + l;
    const int hrow  = l;                 // A-matrix row (M); valid when < H
    const bool hval = (hrow < HH);

    const float* matb  = matrix + (size_t)b * ((size_t)KROWS * DMEM);
    const int    krow0 = ks * (KROWS / KSPLIT);   // 1024 rows per block
    const int    krow1 = krow0 + (KROWS / KSPLIT);

    v8f c = {};
    for (int kr = krow0; kr < krow1; kr += 4) {
        const int m  = kr >> 6;                       // same m for all 4 rows
        const int a0 = (kr & 63) + half * 2;
        v2f A, Bf;
        const float wm = hval ? s_w[hrow * MM + m] : 0.0f;
        A.x = hval ? wm * s_rq[hrow * DADDR + a0]     : 0.0f;
        A.y = hval ? wm * s_rq[hrow * DADDR + a0 + 1] : 0.0f;

        const float* p = matb + (size_t)(kr + half * 2) * DMEM + dcol;
        Bf.x = p[0];
        Bf.y = p[DMEM];

        if ((kr & 63) == 0 && kr + 64 < krow1) {      // prefetch one m-row ahead
            __builtin_prefetch(p + (size_t)64 * DMEM, 0, 1);
        }

        c = __builtin_amdgcn_wmma_f32_16x16x4_f32(false, A, false, Bf,
                                                  (short)0, c, false, false);
    }

    // D layout: vgpr r -> M = r + half*8. Only M < 8 carries data (lanes 0-15).
    if (half == 0) {
        float* outb = num_partial + (size_t)b * HH * DMEM;
#pragma unroll
        for (int r = 0; r < 8; ++r) {
            unsafeAtomicAdd(&outb[r * DMEM + dcol], c[r]);
        }
    }
}

// ---------------------------------------------------------------------------
// K6: den2[b,h] = (sum_m w*q)/den1 + EPS ;  response = num_hat/(den1*den2)
// grid = B blocks of 128 threads (one per d)
// ---------------------------------------------------------------------------
__global__ __launch_bounds__(128) void rm_response_kernel(const float* __restrict__ num_partial,
                                                          const float* __restrict__ w,
                                                          const float* __restrict__ q,
                                                          const float* __restrict__ den1,
                                                          float* __restrict__ response) {
    __shared__ float s_den[HH];          // den1*den2 per head
    const int b   = blockIdx.x;
    const int tid = threadIdx.x;
    if (tid < HH) {
        const int bh = b * HH + tid;
        const float* wr = w + (size_t)bh * MM;
        const float* qr = q + (size_t)bh * MM;
        float s = 0.0f;
        for (int m = 0; m < MM; ++m) s = fmaf(wr[m], qr[m], s);
        const float d1 = den1[bh];
        s_den[tid] = d1 * (s / d1 + EPSF);
    }
    __syncthreads();
#pragma unroll
    for (int h = 0; h < HH; ++h) {
        const float v = num_partial[((size_t)b * HH + h) * DMEM + tid];
        response[(size_t)b * (HH * DMEM) + h * DMEM + tid] = v / s_den[h];
    }
}

// ---------------------------------------------------------------------------
// K7: out = response (64x1024) @ Wm (1024x1024) + bm, WMMA f32 16x16x4
// grid = 4 Mtiles * 64 Ntiles = 256 blocks of 1 wave
// ---------------------------------------------------------------------------
__global__ __launch_bounds__(32) void rm_outgemm_kernel(const float* __restrict__ response,
                                                        const float* __restrict__ Wm,
                                                        const float* __restrict__ bm,
                                                        float* __restrict__ out) {
    const int tile  = blockIdx.x;
    const int Mbase = (tile >> 6) * 16;
    const int Nbase = (tile & 63) * 16;
    const int lane  = threadIdx.x;
    const int half  = lane >> 4;
    const int l     = lane & 15;
    const int row   = Mbase + l;
    const int col   = Nbase + l;

    v8f c = {};
    for (int kk = 0; kk < DMODEL; kk += 4) {
        const int ka = kk + half * 2;
        v2f A, Bf;
        A.x  = response[row * DMODEL + ka];
        A.y  = response[row * DMODEL + ka + 1];
        Bf.x = Wm[(size_t)ka * DMODEL + col];
        Bf.y = Wm[(size_t)(ka + 1) * DMODEL + col];
        c = __builtin_amdgcn_wmma_f32_16x16x4_f32(false, A, false, Bf,
                                                  (short)0, c, false, false);
    }
#pragma unroll
    for (int r = 0; r < 8; ++r) {
        const int mrow = Mbase + r + half * 8;
        out[mrow * DMODEL + col] = c[r] + bm[col];
    }
}

// ---------------------------------------------------------------------------
// launch
// ---------------------------------------------------------------------------
extern "C" void kernel_launch(void* const* d_in, const int* in_sizes, int n_in,
                              void* d_out, int out_size, void* d_ws, size_t ws_size,
                              hipStream_t stream) {
    const float* query      = (const float*)d_in[0];
    const float* addresses  = (const float*)d_in[1];
    const float* matrix     = (const float*)d_in[2];
    const float* normalizer = (const float*)d_in[3];
    const float* Wa         = (const float*)d_in[4];
    const float* ba         = (const float*)d_in[5];
    const float* Wr         = (const float*)d_in[6];
    const float* br         = (const float*)d_in[7];
    const float* Wm         = (const float*)d_in[8];
    const float* bm         = (const float*)d_in[9];
    float* out = (float*)d_out;

    // workspace layout (floats)
    float* ws       = (float*)d_ws;
    float* kbuf     = ws;                           // 8192
    float* norm_k   = kbuf + MM * DADDR;            // 64
    float* aq       = norm_k + DADDR;               // 32768
    float* rq       = aq + BB * NPROJ;              // 32768
    float* wbuf     = rq + BB * NPROJ;              // 65536
    float* qbuf     = wbuf + BB * HH * MM;          // 65536
    float* den1     = qbuf + BB * HH * MM;          // 512
    float* num_part = den1 + BB * HH;               // 65536
    float* response = num_part + BB * HH * DMEM;    // 65536

    rm_kaddr_kernel<<<1, 64, 0, stream>>>(addresses, kbuf, norm_k);

    rm_proj_kernel<<<256, 32, 0, stream>>>(query, Wa, ba, Wr, br, aq, rq);

    rm_weights_kernel<<<BB * HH, 128, 0, stream>>>(aq, rq, kbuf, norm_k,
                                                   normalizer, wbuf, qbuf, den1);

    const int nacc = BB * HH * DMEM;
    rm_zero_kernel<<<(nacc + 255) / 256, 256, 0, stream>>>(num_part, nacc);

    rm_stage1_kernel<<<dim3(BB, KSPLIT), 256, 0, stream>>>(matrix, wbuf, rq, num_part);

    rm_response_kernel<<<BB, 128, 0, stream>>>(num_part, wbuf, qbuf, den1, response);

    rm_outgemm_kernel<<<256, 32, 0, stream>>>(response, Wm, bm, out);
}